// MambaBlock_1133871366249
// MI455X (gfx1250) — compile-verified
//
#include <hip/hip_runtime.h>

// ---------------------------------------------------------------------------
// Mamba block for MI455X (gfx1250), fp32.
//   K1: fused WMMA GEMM (skips unused proj quarter) + softplus/elementwise
//       + per-chunk linear-recurrence composition (P,Q)  [one block == one chunk]
//   K2: cross-chunk exclusive prefix of states (tiny, sequential over chunks).
//   K3: within-chunk replay + output elementwise (re-reads hit L2: 128MB < 192MB).
// ---------------------------------------------------------------------------

typedef __attribute__((ext_vector_type(2))) float v2f;
typedef __attribute__((ext_vector_type(8))) float v8f;

#define H      128
#define FOURH  512
#define LEN    8192
#define BATCH  8
#define MROWS  (BATCH * LEN)        // 65536 GEMM rows
#define NCHUNK 64
#define TCH    (LEN / NCHUNK)       // 128 steps per chunk == rows per k_proj block
#define NSEG   3                    // delta, C, residual (skip unused segment)
#define LDSROW 132                  // padded LDS row stride (floats): bank stride 4

// ---------------- Kernel 1: WMMA GEMM + elementwise + chunk reduce ----------
// Block: 256 threads (8 waves). Block blk covers batch blk/64, chunk blk%64,
// i.e. rows [blk*128, +128). Wave w owns rows [blockRow + 16w, +16).
__global__ __launch_bounds__(256)
void k_proj(const float* __restrict__ x, const float* __restrict__ W,
            const float* __restrict__ Avec, const float* __restrict__ Bvec,
            const float* __restrict__ Dvec,
            float* __restrict__ dA, float* __restrict__ dB,
            float* __restrict__ Cc, float* __restrict__ rD,
            float* __restrict__ Ap, float* __restrict__ Bq)
{
    extern __shared__ float lds[];            // NSEG*H rows x LDSROW floats
    const int tid = threadIdx.x;

    // Stage W rows p in [0,128) U [256,512), compacted to q in [0,384),
    // row-major with padded stride (conflict-free B reads: bank stride 4).
    for (int idx = tid; idx < (NSEG * H * H) / 4; idx += 256) {
        const int q  = idx >> 5;              // compact row (H/4 = 32 float4/row)
        const int k4 = (idx & 31) << 2;
        const int p  = q + ((q >= H) ? H : 0);  // global W row
        const float4 w = *(const float4*)(W + p * H + k4);
        *(float4*)(lds + q * LDSROW + k4) = w;
    }
    __syncthreads();

    const int wave = tid >> 5;
    const int lane = tid & 31;
    const int lm   = lane & 15;               // M (for A) / N (for B) sub-index
    const int kh   = lane >> 4;               // K half-select
    const long rowBase = (long)blockIdx.x * 128 + wave * 16;

    // Preload A operand: 16 rows x K=128 -> 32 v2f (64 VGPRs), reused 24x.
    // V_WMMA_F32_16X16X4_F32 A layout: lane<16 holds K={k,k+1}, lane>=16 K={k+2,k+3}.
    v2f a[32];
    {
        const float* xp = x + (rowBase + lm) * H + 2 * kh;
        #pragma unroll
        for (int kt = 0; kt < 32; ++kt)
            a[kt] = *(const v2f*)(xp + kt * 4);
    }

    // Per-lane chunk-composition pieces: for each owned column h (8 of them:
    // 2 cc x 4 nt), (P,Q) composed over this lane's 8 consecutive time-steps.
    v2f comp[8];

    for (int cc = 0; cc < 2; ++cc) {          // two 64-wide column halves
        v8f accD[4], accC[4], accR[4];
        #pragma unroll
        for (int nt = 0; nt < 4; ++nt) {
            accD[nt] = (v8f){};
            accC[nt] = (v8f){};
            accR[nt] = (v8f){};
        }

        #pragma unroll
        for (int kt = 0; kt < 32; ++kt) {
            const int kofs = kt * 4 + 2 * kh;
            #pragma unroll
            for (int nt = 0; nt < 4; ++nt) {
                const int c = cc * 64 + nt * 16 + lm;    // column within segment
                const v2f bD = *(const v2f*)(lds + (0 * H + c) * LDSROW + kofs);
                const v2f bC = *(const v2f*)(lds + (1 * H + c) * LDSROW + kofs);
                const v2f bR = *(const v2f*)(lds + (2 * H + c) * LDSROW + kofs);
                accD[nt] = __builtin_amdgcn_wmma_f32_16x16x4_f32(
                    false, a[kt], false, bD, (short)0, accD[nt], false, false);
                accC[nt] = __builtin_amdgcn_wmma_f32_16x16x4_f32(
                    false, a[kt], false, bC, (short)0, accC[nt], false, false);
                accR[nt] = __builtin_amdgcn_wmma_f32_16x16x4_f32(
                    false, a[kt], false, bR, (short)0, accR[nt], false, false);
            }
        }

        // Elementwise + store + local (P,Q) composition.
        // C/D layout: VGPR r -> row r (lanes 0-15) / row r+8 (lanes 16-31),
        // so this lane's rows are l = wave*16 + 8*kh + r, r = 0..7 (increasing l).
        #pragma unroll
        for (int nt = 0; nt < 4; ++nt) {
            const int h  = cc * 64 + nt * 16 + lm;
            const float Ah = Avec[h];
            const float Bh = Bvec[h];
            const float Dh = Dvec[h];
            float P = 1.f, Q = 0.f;
            #pragma unroll
            for (int r = 0; r < 8; ++r) {
                const long row = rowBase + r + 8 * kh;
                const float dv = accD[nt][r];
                const float sp = fmaxf(dv, 0.f) + __logf(1.f + __expf(-fabsf(dv)));
                const float rs = accR[nt][r];
                const float av = sp * Ah;
                const float bv = sp * Bh * rs;
                const long o = row * H + h;
                dA[o] = av;
                dB[o] = bv;
                Cc[o] = accC[nt][r];
                rD[o] = rs * Dh;
                Q = av * Q + bv;              // compose step l: s' = a*s + b
                P = av * P;
            }
            comp[cc * 4 + nt] = (v2f){P, Q};
        }
    }

    // ---- Combine 16 per-lane pieces per h (order: wave-major, then kh) ----
    __syncthreads();                          // done with W region of LDS
    {
        v2f* pieces = (v2f*)lds;              // [h][16] = 2048 v2f = 16KB
        #pragma unroll
        for (int i = 0; i < 8; ++i) {
            const int h = (i >> 2) * 64 + (i & 3) * 16 + lm;
            pieces[h * 16 + wave * 2 + kh] = comp[i];
        }
    }
    __syncthreads();
    if (tid < H) {
        const v2f* pieces = (const v2f*)lds;
        float P = 1.f, Q = 0.f;
        #pragma unroll
        for (int i = 0; i < 16; ++i) {        // i = wave*2 + kh: increasing l
            const v2f pq = pieces[tid * 16 + i];
            Q = pq.x * Q + pq.y;
            P = pq.x * P;
        }
        const long o = (long)blockIdx.x * H + tid;   // (b*NCHUNK + c)*H + h
        Ap[o] = P;
        Bq[o] = Q;
    }
}

// ---------------- Kernel 2: cross-chunk exclusive prefix --------------------
__global__ __launch_bounds__(128)
void k_chunk_scan(const float* __restrict__ Ap, const float* __restrict__ Bq,
                  float* __restrict__ S)
{
    const int h = threadIdx.x, b = blockIdx.x;
    float s = 0.f;
    for (int c = 0; c < NCHUNK; ++c) {
        const long o = (long)(b * NCHUNK + c) * H + h;
        S[o] = s;                 // state entering chunk c
        s = Ap[o] * s + Bq[o];
    }
}

// ---------------- Kernel 3: replay + output ---------------------------------
__global__ __launch_bounds__(128)
void k_apply(const float* __restrict__ dA, const float* __restrict__ dB,
             const float* __restrict__ Cc, const float* __restrict__ rD,
             const float* __restrict__ S, float* __restrict__ out)
{
    const int h = threadIdx.x;
    const int c = blockIdx.x, b = blockIdx.y;
    float s = S[(long)(b * NCHUNK + c) * H + h];
    const long base = ((long)(b * LEN + c * TCH)) * H + h;
    for (int t = 0; t < TCH; ++t) {
        const long o = base + (long)t * H;
        s = dA[o] * s + dB[o];
        out[o] = s * Cc[o] + rD[o];
    }
}

// ---------------------------------------------------------------------------
extern "C" void kernel_launch(void* const* d_in, const int* in_sizes, int n_in,
                              void* d_out, int out_size, void* d_ws, size_t ws_size,
                              hipStream_t stream)
{
    (void)in_sizes; (void)n_in; (void)out_size; (void)ws_size;

    const float* x  = (const float*)d_in[0];   // [B, L, H]
    const float* W  = (const float*)d_in[1];   // [4H, H]
    const float* Av = (const float*)d_in[2];   // [H, 1]
    const float* Bv = (const float*)d_in[3];   // [H, 1]
    const float* Dv = (const float*)d_in[4];   // [H]
    float* out = (float*)d_out;                // [B, L, H]

    const size_t NE = (size_t)BATCH * LEN * H; // 8388608
    const size_t NC = (size_t)BATCH * NCHUNK * H;
    float* ws = (float*)d_ws;
    float* dA = ws;            // NE
    float* dB = dA + NE;       // NE
    float* Cc = dB + NE;       // NE
    float* rD = Cc + NE;       // NE
    float* Ap = rD + NE;       // NC
    float* Bq = Ap + NC;       // NC
    float* S  = Bq + NC;       // NC

    const int ldsBytes = NSEG * H * LDSROW * (int)sizeof(float);  // 202752 B
    hipFuncSetAttribute(reinterpret_cast<const void*>(k_proj),
                        hipFuncAttributeMaxDynamicSharedMemorySize, ldsBytes);

    k_proj<<<MROWS / 128, 256, ldsBytes, stream>>>(x, W, Av, Bv, Dv,
                                                   dA, dB, Cc, rD, Ap, Bq);
    k_chunk_scan<<<BATCH, 128, 0, stream>>>(Ap, Bq, S);
    k_apply<<<dim3(NCHUNK, BATCH), 128, 0, stream>>>(dA, dB, Cc, rD, S, out);
}